// Dynamic_conv2d_v2_88261577933065
// MI455X (gfx1250) — compile-verified
//
#include <hip/hip_runtime.h>

typedef __attribute__((ext_vector_type(2))) float v2f;
typedef __attribute__((ext_vector_type(8))) float v8f;
typedef __attribute__((ext_vector_type(4))) unsigned int u32x4;
typedef __attribute__((ext_vector_type(8))) int i32x8;
typedef __attribute__((ext_vector_type(4))) int i32x4;

#define B_    8
#define CIN_  64
#define COUT_ 64
#define H_    256
#define W_    256
#define KG    (CIN_ * 9)   // 576 = GEMM K (tap-major: kk = tap*64 + cin)
#define NT    32           // pixels (N columns) per block strip
#define XRS   34           // slab row stride in floats (cols w0-1 .. w0+32)
#define RPB   16           // output rows per block
#define SLABF (CIN_ * XRS) // floats per row-slab (2176)

// ---------------------------------------------------------------------------
// Kernel 1: aggregated per-batch weights, K-pair-interleaved:
//   agg[ b ][ kk>>1 ][ cout ][ kk&1 ],  kk = tap*64 + cin
// so a WMMA A-fragment (two adjacent K values for one cout) is ONE b64 load.
// ---------------------------------------------------------------------------
__global__ __launch_bounds__(256) void agg_weights_kernel(
    const float* __restrict__ att, const float* __restrict__ weight,
    const float* __restrict__ tb,  const float* __restrict__ tq,
    const float* __restrict__ tn,  const float* __restrict__ tx,
    const float* __restrict__ mb,  const float* __restrict__ mq,
    const float* __restrict__ mn,  const float* __restrict__ mx,
    float* __restrict__ agg)
{
    int e = blockIdx.x * 256 + threadIdx.x;       // e = ((b*KG + kk)*COUT + m)
    if (e >= B_ * KG * COUT_) return;
    int m   = e & (COUT_ - 1);
    int kk  = (e >> 6) % KG;
    int b   = e / (KG * COUT_);
    int cin = kk & (CIN_ - 1);
    int tap = kk >> 6;
    int wi  = (m * CIN_ + cin) * 9 + tap;         // OIHW index
    int mi  = m * CIN_ + cin;
    float a0 = att[b * 10 + 0];
    float a1 = att[b * 10 + 1];
    float a2 = att[b * 10 + 2];
    float v = weight[wi]
            + tb[wi] * mb[mi] * a1
            + tq[wi] * mq[mi] * a2
            + tn[wi] * mn[mi] * a0
            + tx[wi] * mx[mi] * a0;
    agg[((size_t)(b * (KG / 2) + (kk >> 1)) * COUT_ + m) * 2 + (kk & 1)] = v;
}

// Cooperative (scalar) slab load with zero padding: row r, cols w0-1..w0+32.
__device__ __forceinline__ void load_slab_manual(
    float* xs, const float* xb, int r, int w0, int tid)
{
    unsigned sb = (unsigned)(r & 3) * SLABF;
    for (int idx = tid; idx < SLABF; idx += 256) {
        int cin = idx / XRS;
        int c   = idx - cin * XRS;
        int col = w0 - 1 + c;
        float v = 0.0f;
        if (r >= 0 && r < H_ && col >= 0 && col < W_)
            v = xb[(size_t)cin * (H_ * W_) + (size_t)r * W_ + col];
        xs[sb + idx] = v;
    }
}

// ---------------------------------------------------------------------------
// Kernel 2: persistent-row implicit-GEMM conv with V_WMMA_F32_16X16X4_F32 and
// TDM (tensor_load_to_lds) software pipelining.
// Block = (batch b, 16-row band, 32-pixel strip). Ring of 4 row-slabs in LDS;
// wave 0 issues an async TDM load of row h+2 while all waves compute row h.
// ---------------------------------------------------------------------------
__global__ __launch_bounds__(256) void dynconv_wmma_kernel(
    const float* __restrict__ x, const float* __restrict__ agg,
    const float* __restrict__ bias, float* __restrict__ out)
{
    __shared__ float xs[4 * SLABF];               // 4 slabs * 2176 f = 34,816 B

    const int strips = W_ / NT;                   // 8
    const int hblks  = H_ / RPB;                  // 16
    int bid = blockIdx.x;
    int b   = bid / (hblks * strips);
    int rem = bid % (hblks * strips);
    int h0  = (rem / strips) * RPB;
    int w0  = (rem % strips) * NT;
    const bool interior = (w0 != 0) && (w0 + NT != W_);   // full halo in-range

    int tid = threadIdx.x;
    const float* xb = x + (size_t)b * CIN_ * H_ * W_;

    // ---- prologue: slabs for rows h0-1, h0, h0+1 ---------------------------
    load_slab_manual(xs, xb, h0 - 1, w0, tid);
    load_slab_manual(xs, xb, h0,     w0, tid);
    load_slab_manual(xs, xb, h0 + 1, w0, tid);
    __syncthreads();

    int wid   = tid >> 5;
    int lane  = tid & 31;
    int l16   = lane & 15;
    int hiSel = (lane >> 4) & 1;                  // lane half selects K pair
    int m0    = (wid & 3) * 16;
    int n0    = (wid >> 2) * 16;

    const float* aggB = agg + (size_t)b * KG * COUT_;   // pair-interleaved
    int mrow = m0 + l16;
    int ncol = n0 + l16;

    float biasv[8];
#pragma unroll
    for (int r = 0; r < 8; ++r) biasv[r] = bias[m0 + r + 8 * hiSel];

    float* ob = out + (size_t)b * COUT_ * H_ * W_ + w0 + ncol;

    // constant parts of the TDM descriptor (D#), ISA ch.8:
    // group1: data_size=4B(code 2), tensor_dim0=34, tensor_dim1=64,
    //         tile_dim0=34, tile_dim1=64, tensor_dim0_stride=H*W
    i32x8 g1;
    g1[0] = 2 << 16;                 // data_size=2 (4 bytes); mask/flags = 0
    g1[1] = XRS << 16;               // tensor_dim0[15:0]
    g1[2] = CIN_ << 16;              // tensor_dim0[31:16]=0 | tensor_dim1[15:0]
    g1[3] = XRS << 16;               // tensor_dim1[31:16]=0 | tile_dim0
    g1[4] = CIN_;                    // tile_dim1 | tile_dim2=0
    g1[5] = H_ * W_;                 // tensor_dim0_stride[31:0]
    g1[6] = 0;                       // stride[47:32] | tensor_dim1_stride lo
    g1[7] = 0;
    i32x4 gz  = {0, 0, 0, 0};        // groups 2/3 unused (2D tile)
    i32x8 gz8 = {0, 0, 0, 0, 0, 0, 0, 0};

    const int hLast = h0 + RPB - 1;
    for (int h = h0; h <= hLast; ++h) {
        // ---- prefetch slab for row h+2 (overlapped with compute) ----------
        bool tdmThis = false;
        if (h < hLast) {
            int r = h + 2;
            if (interior && r < H_) {
                tdmThis = true;
                if (wid == 0) {
                    unsigned sb = (unsigned)(r & 3) * SLABF;
                    unsigned ldsAddr = (unsigned)(uintptr_t)(xs + sb);
                    unsigned long long ga =
                        (unsigned long long)(uintptr_t)(xb + (size_t)r * W_ + (w0 - 1));
                    u32x4 g0;
                    g0[0] = 1u;                               // count=1, flags=0
                    g0[1] = ldsAddr;                          // lds_addr (bytes)
                    g0[2] = (unsigned)(ga & 0xFFFFFFFFu);     // global_addr[31:0]
                    g0[3] = (unsigned)((ga >> 32) & 0x01FFFFFFu) | (2u << 30); // addr hi | type=2
                    __builtin_amdgcn_tensor_load_to_lds(g0, g1, gz, gz, gz8, 0);
                }
            } else {
                load_slab_manual(xs, xb, r, w0, tid);   // edges / beyond-bottom
            }
        }

        // ---- compute output row h: M=64 x N=32 x K=576 GEMM ---------------
        v8f acc = {};
        for (int tap = 0; tap < 9; ++tap) {
            int kh = tap / 3, kw = tap % 3;
            const float* aBase = aggB + ((size_t)(tap * 32) * COUT_ + mrow) * 2;
            const float* bBase = xs + (unsigned)((h - 1 + kh) & 3) * SLABF + ncol + kw;
#pragma unroll 8
            for (int s = 0; s < 16; ++s) {
                int cp = 2 * s + hiSel;               // cin-pair index
                v2f af = *(const v2f*)(aBase + (size_t)cp * (COUT_ * 2));
                v2f bf;
                bf.x = bBase[(2 * cp + 0) * XRS];
                bf.y = bBase[(2 * cp + 1) * XRS];
                acc = __builtin_amdgcn_wmma_f32_16x16x4_f32(
                    false, af, false, bf, (short)0, acc, false, false);
            }
        }

        // ---- epilogue: D VGPR r -> M = m0 + r + 8*hiSel, N = n0 + l16 -----
#pragma unroll
        for (int r = 0; r < 8; ++r) {
            int m = m0 + r + 8 * hiSel;
            ob[(size_t)m * (H_ * W_) + (size_t)h * W_] = acc[r] + biasv[r];
        }

        // ---- close the pipeline stage -------------------------------------
        if (tdmThis && wid == 0)
            __builtin_amdgcn_s_wait_tensorcnt(0);     // TDM data landed in LDS
        __syncthreads();                              // publish slab h+2
    }
}

// ---------------------------------------------------------------------------
extern "C" void kernel_launch(void* const* d_in, const int* in_sizes, int n_in,
                              void* d_out, int out_size, void* d_ws, size_t ws_size,
                              hipStream_t stream)
{
    const float* x      = (const float*)d_in[0];
    const float* att    = (const float*)d_in[1];
    const float* weight = (const float*)d_in[2];
    const float* tb     = (const float*)d_in[3];
    const float* tq     = (const float*)d_in[4];
    const float* tn     = (const float*)d_in[5];
    const float* tx     = (const float*)d_in[6];
    const float* mb     = (const float*)d_in[7];
    const float* mq     = (const float*)d_in[8];
    const float* mn     = (const float*)d_in[9];
    const float* mx     = (const float*)d_in[10];
    const float* bias   = (const float*)d_in[11];
    float* out = (float*)d_out;
    float* agg = (float*)d_ws;                    // 8*576*64*4 = 1.18 MB

    int n_agg = B_ * KG * COUT_;
    agg_weights_kernel<<<(n_agg + 255) / 256, 256, 0, stream>>>(
        att, weight, tb, tq, tn, tx, mb, mq, mn, mx, agg);

    int nblocks = B_ * (H_ / RPB) * (W_ / NT);    // 1024
    dynconv_wmma_kernel<<<nblocks, 256, 0, stream>>>(x, agg, bias, out);
}